// ClipPairWiseLossAll_26139170964397
// MI455X (gfx1250) — compile-verified
//
#include <hip/hip_runtime.h>
#include <hip/hip_bf16.h>
#include <math.h>

// Problem constants (from reference: M=512, N=768)
#define MM 512
#define NN 768
#define BI 16            // i-rows per block tile
#define BJ 16            // j-rows per block tile (one wave per j)
#define PAD 772          // LDS row stride in floats (768+4) -> conflict-free banks
#define NBX 32           // MM / BI
#define NBLK (NBX * NBX) // partials in workspace

typedef float v2f __attribute__((ext_vector_type(2)));
typedef float v8f __attribute__((ext_vector_type(8)));

// Kernel 1: one block = 16x16 pair tile. 512 threads = 16 waves (wave32).
// Wave w computes pairs (i0+p, j0+w) for p=0..15 using WMMA f32 16x16x4:
//   A[p][k] = relu(r_i[k]-r_j[k])^2, B = ones  =>  D[p][*] += sum_k A[p][k]
__global__ __launch_bounds__(512)
void cplw_pairs_kernel(const float* __restrict__ repr,
                       const int*  __restrict__ GT,
                       float*      __restrict__ partials) {
    __shared__ float ldsI[BI * PAD];   // 16 gathered i-rows, padded
    __shared__ float red[2 * BJ];      // per-(wave,half) partial norms

    const int tid = threadIdx.x;
    const int bi = blockIdx.y;
    const int bj = blockIdx.x;
    const int blk = bi * NBX + bj;

    if (bj < bi) {                     // strictly-lower tile: no i<j pairs
        if (tid == 0) partials[blk] = 0.0f;
        return;
    }

    const int i0 = bi * BI;
    const int j0 = bj * BJ;

    // ---- Stage the 16 i-rows (gathered through GT) into LDS ----
    for (int idx = tid; idx < BI * NN; idx += 512) {
        const int r = idx / NN;
        const int c = idx - r * NN;
        const int row = GT[i0 + r];
        ldsI[r * PAD + c] = repr[(size_t)row * NN + c];
    }
    __syncthreads();

    const int wv   = tid >> 5;         // wave id = local j index
    const int lane = tid & 31;
    const int p    = lane & 15;        // pair row (A-matrix M index)
    const int half = lane >> 4;        // selects K=0,1 vs K=2,3 slots

    const int j    = j0 + wv;
    const int rowj = GT[j];
    const float* __restrict__ rj = repr + (size_t)rowj * NN;

    const float* __restrict__ ri = &ldsI[p * PAD];

    v8f acc = {0.f, 0.f, 0.f, 0.f, 0.f, 0.f, 0.f, 0.f};
    const v2f ones = {1.0f, 1.0f};

    // ---- K loop: 192 WMMA steps, 4 features each ----
    #pragma unroll 4
    for (int k = 0; k < NN; k += 4) {
        const int k0 = k + 2 * half;                       // A layout: VGPR0<-k0, VGPR1<-k0+1
        const v2f ai = *reinterpret_cast<const v2f*>(ri + k0);   // ds_load_b64, bank-clean
        const v2f bv = *reinterpret_cast<const v2f*>(rj + k0);   // lane-uniform global b64
        const float d0 = fmaxf(ai.x - bv.x, 0.0f);
        const float d1 = fmaxf(ai.y - bv.y, 0.0f);
        v2f a;
        a.x = d0 * d0;
        a.y = d1 * d1;
        // D = A * ones + C  -> every column of row p accumulates sum_k relu^2
        acc = __builtin_amdgcn_wmma_f32_16x16x4_f32(
            /*neg_a=*/false, a, /*neg_b=*/false, ones,
            /*c_mod=*/(short)0, acc, /*reuse_a=*/false, /*reuse_b=*/false);
    }

    // ---- Extract diagonal: lane0 holds rows 0..7, lane16 rows 8..15 ----
    if (p == 0) {
        float s = 0.0f;
        #pragma unroll
        for (int v = 0; v < 8; ++v) {
            const int i = i0 + half * 8 + v;               // C layout: M = v + 8*half
            if (i < j) s += sqrtf(fmaxf(acc[v], 0.0f));
        }
        red[wv * 2 + half] = s;
    }
    __syncthreads();

    if (tid == 0) {
        float t = 0.0f;
        #pragma unroll
        for (int q = 0; q < 2 * BJ; ++q) t += red[q];
        partials[blk] = t;
    }
}

// Kernel 2: deterministic fixed-order reduction of block partials.
__global__ __launch_bounds__(256)
void cplw_reduce_kernel(const float* __restrict__ partials,
                        float* __restrict__ out) {
    __shared__ float sb[256];
    const int tid = threadIdx.x;
    float s = 0.0f;
    for (int idx = tid; idx < NBLK; idx += 256) s += partials[idx];
    sb[tid] = s;
    __syncthreads();
    if (tid == 0) {
        float t = 0.0f;
        for (int q = 0; q < 256; ++q) t += sb[q];
        out[0] = t;
    }
}

extern "C" void kernel_launch(void* const* d_in, const int* in_sizes, int n_in,
                              void* d_out, int out_size, void* d_ws, size_t ws_size,
                              hipStream_t stream) {
    const float* repr = (const float*)d_in[0];
    const int*   GT   = (const int*)d_in[1];
    float* out      = (float*)d_out;
    float* partials = (float*)d_ws;    // NBLK floats = 4 KB

    dim3 grid(NBX, NBX);
    cplw_pairs_kernel<<<grid, 512, 0, stream>>>(repr, GT, partials);
    cplw_reduce_kernel<<<1, 256, 0, stream>>>(partials, out);
}